// K_WTA2D_6425271075428
// MI455X (gfx1250) — compile-verified
//
#include <hip/hip_runtime.h>
#include <stdint.h>

// WTA-2D sparsity: per (B,C) row of 3136 floats, zero everything >= the
// k-th largest (k = 313), keep the rest. Exact radix-select in LDS.
//
// CDNA5 path: rows staged global -> LDS with global_load_async_to_lds_b128
// (ASYNCcnt), one HBM read + one non-temporal HBM write per element.
// All LDS sweeps are vectorized to b128 (3136 floats = 784 x uint4).

#define LAYER    3136   // 56*56 elements per row (fixed by reference)
#define NCHUNK   784    // LAYER / 4 uint4 chunks (= 16-byte async chunks)
#define NTHREADS 256

typedef uint32_t vuint4  __attribute__((ext_vector_type(4)));
typedef float    vfloat4 __attribute__((ext_vector_type(4)));

__device__ __forceinline__ uint32_t monotone(uint32_t b) {
    // unsigned order of result == float order (ascending)
    return b ^ ((b & 0x80000000u) ? 0xFFFFFFFFu : 0x80000000u);
}

__global__ __launch_bounds__(NTHREADS)
void K_WTA2D_kernel(const float* __restrict__ x, float* __restrict__ out, int k) {
    __shared__ __align__(16) uint32_t ldsData[LAYER];
    __shared__ uint32_t hist[256];
    __shared__ uint32_t scanb[256];
    __shared__ uint32_t selInfo[2];

    const int tid = threadIdx.x;
    const size_t rowBase = (size_t)blockIdx.x * LAYER;
    const float* src = x + rowBase;

    // ---- Stage the row into LDS via CDNA5 async global->LDS copies ----
    {
        uint32_t ldsBase = (uint32_t)(uintptr_t)(&ldsData[0]);   // LDS byte offset
        uint64_t gBase   = (uint64_t)(uintptr_t)src;
        for (int c = tid; c < NCHUNK; c += NTHREADS) {
            uint32_t ldsOff = ldsBase + (uint32_t)c * 16u;
            uint64_t gaddr  = gBase + (uint64_t)c * 16u;
            asm volatile("global_load_async_to_lds_b128 %0, %1, off"
                         :: "v"(ldsOff), "v"(gaddr) : "memory");
        }
        // wait for this wave's async copies to land in LDS
        asm volatile("s_wait_asynccnt 0x0" ::: "memory");
    }
    __syncthreads();   // all waves' chunks visible

    const vuint4* ldsVec = (const vuint4*)ldsData;

    // ---- Exact k-th-largest via MSB-first radix select (4 x 8-bit) ----
    uint32_t prefixVal  = 0u;
    uint32_t prefixMask = 0u;
    uint32_t kk = (uint32_t)k;   // rank from the top, 1-indexed

    for (int pass = 0; pass < 4; ++pass) {
        const int shift = 24 - 8 * pass;

        hist[tid] = 0u;
        __syncthreads();

        for (int c = tid; c < NCHUNK; c += NTHREADS) {
            vuint4 q = ldsVec[c];                     // ds_load_b128
            uint32_t u0 = monotone(q.x), u1 = monotone(q.y);
            uint32_t u2 = monotone(q.z), u3 = monotone(q.w);
            if ((u0 & prefixMask) == prefixVal) atomicAdd(&hist[(u0 >> shift) & 0xFFu], 1u);
            if ((u1 & prefixMask) == prefixVal) atomicAdd(&hist[(u1 >> shift) & 0xFFu], 1u);
            if ((u2 & prefixMask) == prefixVal) atomicAdd(&hist[(u2 >> shift) & 0xFFu], 1u);
            if ((u3 & prefixMask) == prefixVal) atomicAdd(&hist[(u3 >> shift) & 0xFFu], 1u);
        }
        __syncthreads();

        // Inclusive suffix sum over the 256 bins (Hillis-Steele).
        uint32_t v = hist[tid];
        scanb[tid] = v;
        __syncthreads();
        for (int d = 1; d < 256; d <<= 1) {
            uint32_t add = (tid + d < 256) ? scanb[tid + d] : 0u;
            __syncthreads();
            scanb[tid] += add;
            __syncthreads();
        }

        // Exactly one bin satisfies: (#candidates in larger bins) < kk <= suffix sum.
        uint32_t S = scanb[tid];          // candidates with byte >= tid
        uint32_t cumGreater = S - v;      // candidates with byte >  tid
        if (S >= kk && cumGreater < kk) {
            selInfo[0] = (uint32_t)tid;
            selInfo[1] = kk - cumGreater; // rank within the selected bin
        }
        __syncthreads();

        uint32_t sel = selInfo[0];
        kk = selInfo[1];
        prefixVal  |= sel << shift;
        prefixMask |= 0xFFu << shift;
        __syncthreads();
    }

    const uint32_t uTop = prefixVal;   // transformed bits of the k-th largest

    // ---- Apply mask (keep strictly-smaller values), b128 non-temporal store ----
    vfloat4* dst = (vfloat4*)(out + rowBase);
    for (int c = tid; c < NCHUNK; c += NTHREADS) {
        vuint4 q = ldsVec[c];                         // ds_load_b128
        vfloat4 r;
        r.x = (monotone(q.x) < uTop) ? __uint_as_float(q.x) : 0.0f;
        r.y = (monotone(q.y) < uTop) ? __uint_as_float(q.y) : 0.0f;
        r.z = (monotone(q.z) < uTop) ? __uint_as_float(q.z) : 0.0f;
        r.w = (monotone(q.w) < uTop) ? __uint_as_float(q.w) : 0.0f;
        __builtin_nontemporal_store(r, dst + c);      // global_store_b128 th:NT
    }
}

extern "C" void kernel_launch(void* const* d_in, const int* in_sizes, int n_in,
                              void* d_out, int out_size, void* d_ws, size_t ws_size,
                              hipStream_t stream) {
    const float* x = (const float*)d_in[0];
    float* out = (float*)d_out;

    const int total = in_sizes[0];          // 64*256*56*56
    const int rows  = total / LAYER;        // 16384 (B*C)
    const int k     = (int)(0.1f * (float)LAYER);   // 313

    K_WTA2D_kernel<<<dim3(rows), dim3(NTHREADS), 0, stream>>>(x, out, k);
}